// Discriminator_70866960384744
// MI455X (gfx1250) — compile-verified
//
#include <hip/hip_runtime.h>
#include <math.h>

#define N_NODES 65536
#define N_EDGES 1048576
#define IN_DIM 256
#define D2 512
#define HID 1024
#define N_GRAPHS 64

typedef float v2f __attribute__((ext_vector_type(2)));
typedef float v8f __attribute__((ext_vector_type(8)));
typedef int   v4i __attribute__((ext_vector_type(4)));

// CDNA5 async global->LDS path (guarded: falls back to plain copies if the
// toolchain doesn't declare the builtin; also keeps the host pass clean).
#if defined(__HIP_DEVICE_COMPILE__) && defined(__has_builtin)
#if __has_builtin(__builtin_amdgcn_global_load_async_to_lds_b128)
#define HAVE_ASYNC_LDS 1
#endif
#endif
#ifndef HAVE_ASYNC_LDS
#define HAVE_ASYNC_LDS 0
#endif

__device__ __forceinline__ void copy16_to_lds(const float* g, float* l) {
#if HAVE_ASYNC_LDS
    // signature (probe-derived): (v4i addrspace(1)*, v4i addrspace(3)*, imm offset, imm cpol)
    __builtin_amdgcn_global_load_async_to_lds_b128(
        (__attribute__((address_space(1))) v4i*)g,
        (__attribute__((address_space(3))) v4i*)l,
        0, 0);
#else
    *reinterpret_cast<float4*>(l) = *reinterpret_cast<const float4*>(g);
#endif
}

__device__ __forceinline__ void async_wait_all() {
#if HAVE_ASYNC_LDS
#if __has_builtin(__builtin_amdgcn_s_wait_asynccnt)
    __builtin_amdgcn_s_wait_asynccnt(0);
#else
    asm volatile("s_wait_asynccnt 0x0" ::: "memory");
#endif
#endif
}

// ---------------- utility zero kernels ----------------
__global__ void zero_f32(float* p, int n) {
    int i = blockIdx.x * 256 + threadIdx.x;
    if (i < n) p[i] = 0.0f;
}
__global__ void zero_i32(int* p, int n) {
    int i = blockIdx.x * 256 + threadIdx.x;
    if (i < n) p[i] = 0;
}

// ---------------- edge degree count ----------------
__global__ void edge_degree(const int* __restrict__ ei, int* __restrict__ deg) {
    int e = blockIdx.x * 256 + threadIdx.x;
    if (e < N_EDGES) atomicAdd(&deg[ei[N_EDGES + e]], 1);
}

// ---------------- per-graph node count ----------------
__global__ void graph_count(const int* __restrict__ batch, int* __restrict__ gcnt) {
    int n = blockIdx.x * 256 + threadIdx.x;
    if (n < N_NODES) atomicAdd(&gcnt[batch[n]], 1);
}

// ---------------- single-block exclusive scan of degrees ----------------
__global__ void scan_deg(const int* __restrict__ deg, int* __restrict__ offs,
                         int* __restrict__ cursor) {
    __shared__ int sums[256];
    const int tid = threadIdx.x;
    const int CHUNK = N_NODES / 256;
    int base = tid * CHUNK;
    int s = 0;
    for (int i = 0; i < CHUNK; i++) s += deg[base + i];
    sums[tid] = s;
    __syncthreads();
    for (int off = 1; off < 256; off <<= 1) {
        int v = (tid >= off) ? sums[tid - off] : 0;
        __syncthreads();
        sums[tid] += v;
        __syncthreads();
    }
    int run = (tid == 0) ? 0 : sums[tid - 1];
    for (int i = 0; i < CHUNK; i++) {
        int d = deg[base + i];
        offs[base + i] = run;
        cursor[base + i] = run;
        run += d;
    }
}

// ---------------- bin edges by destination (counting sort) ----------------
__global__ void bin_edges(const int* __restrict__ ei, int* __restrict__ cursor,
                          int* __restrict__ srcSorted) {
    int e = blockIdx.x * 256 + threadIdx.x;
    if (e < N_EDGES) {
        int d = ei[N_EDGES + e];
        int pos = atomicAdd(&cursor[d], 1);
        srcSorted[pos] = ei[e];
    }
}

// ---------------- neighbor mean aggregation (no atomics) ----------------
__global__ void aggregate_mean(const float* __restrict__ normal,
                               const float* __restrict__ extreme,
                               const int* __restrict__ offs,
                               const int* __restrict__ deg,
                               const int* __restrict__ srcSorted,
                               float* __restrict__ meanOut) {
    int node = blockIdx.x;
    int tid = threadIdx.x;
    int f = tid * 4;
    const float* srcBase = (f < IN_DIM) ? normal : extreme;
    int fo = (f < IN_DIM) ? f : (f - IN_DIM);
    int d = deg[node];
    int start = offs[node];
    float ax = 0.f, ay = 0.f, az = 0.f, aw = 0.f;
    for (int j = 0; j < d; j++) {
        int s = srcSorted[start + j];
        float4 v = *reinterpret_cast<const float4*>(srcBase + (size_t)s * IN_DIM + fo);
        ax += v.x; ay += v.y; az += v.z; aw += v.w;
    }
    float inv = 1.0f / (float)(d > 0 ? d : 1);
    float4 r; r.x = ax * inv; r.y = ay * inv; r.z = az * inv; r.w = aw * inv;
    *reinterpret_cast<float4*>(meanOut + (size_t)node * D2 + f) = r;
}

// ---------------- fused triple GEMM with FP32 WMMA ----------------
// x_combined = relu(mean @ W_l + b_l + x @ W_r) + relu(x @ W_fc1 + b_fc1)
// workgroup tile: 64 rows x 256 cols; 8 waves = 4 row-subtiles x 2 col-halves
// wave tile: 16 rows x 128 cols = 8 coltiles of 16x16, dual accumulators.
// A is double-buffered in LDS and filled by async global->LDS b128 copies
// (ASYNCcnt); B is register-prefetched and stored k-pair-interleaved so each
// WMMA B fragment is one conflict-free ds_load_b64.
#define KB 16
#define A_STRIDE 20        // padded k-stride (floats): conflict-free & 16B aligned
#define BKP 544            // floats per k-pair row: 544 % 64 == 32 -> bank split
#define BMAT (8 * BKP)     // 8 k-pairs per 16-k chunk
#define LA_FLOATS (2 * 64 * A_STRIDE)              // per A buffer (mean + x)
#define LB_FLOATS (3 * BMAT)
#define SMEM_FLOATS (2 * LA_FLOATS + LB_FLOATS)    // 18176 floats = 72704 B

__global__ __launch_bounds__(256) void fused_gemm(
    const float* __restrict__ normal, const float* __restrict__ extreme,
    const float* __restrict__ meanIn,
    const float* __restrict__ Wl, const float* __restrict__ bl,
    const float* __restrict__ Wr,
    const float* __restrict__ Wfc1, const float* __restrict__ bfc1,
    float* __restrict__ xcomb) {
    extern __shared__ float smem[];
    float* lB = smem + 2 * LA_FLOATS;

    const int tid = threadIdx.x;
    const int lane = tid & 31;
    const int wave = tid >> 5;
    const int row_sub = wave >> 1;    // 0..3
    const int col_half = wave & 1;    // 0..1
    const int block_row = blockIdx.x * 64;
    const int block_col = blockIdx.y * 256;

    const int m16 = lane & 15;
    const int kb2 = (lane >> 4) * 2;  // lanes 16..31 handle K+2,K+3

    v8f acc_g[8], acc_m[8];
    {
        v8f zz = {};
        #pragma unroll
        for (int t = 0; t < 8; t++) { acc_g[t] = zz; acc_m[t] = zz; }
    }

    // per-thread staging assignments
    const int a_row = tid >> 2;
    const int a_kq = (tid & 3) * 4;

    // issue A-chunk copies (async on CDNA5): 16B per thread per matrix
    auto issueA = [&](int k0, float* buf) {
        int grow = block_row + a_row;
        const float* gm = meanIn + (size_t)grow * D2 + k0 + a_kq;
        copy16_to_lds(gm, &buf[a_row * A_STRIDE + a_kq]);
        int kk = k0 + a_kq;
        const float* gx = (kk < IN_DIM)
                              ? (normal + (size_t)grow * IN_DIM + kk)
                              : (extreme + (size_t)grow * IN_DIM + (kk - IN_DIM));
        copy16_to_lds(gx, &buf[64 * A_STRIDE + a_row * A_STRIDE + a_kq]);
    };

    // register prefetch buffer for B (interleave transform needs registers)
    float4 pB[12];
    auto loadB = [&](int k0) {
        #pragma unroll
        for (int i = 0; i < 12; i++) {
            int idx = i * 256 + tid;          // 0..3071
            int mat = idx >> 10;              // 0..2
            int rem = idx & 1023;
            int kp = rem >> 7;                // 0..7
            int cp = rem & 127;               // 0..127 column pairs
            const float* W = (mat == 0) ? Wl : ((mat == 1) ? Wr : Wfc1);
            const float* g = W + (size_t)(k0 + kp * 2) * HID + block_col + cp * 2;
            float2 g0 = *reinterpret_cast<const float2*>(g);
            float2 g1 = *reinterpret_cast<const float2*>(g + HID);
            pB[i] = make_float4(g0.x, g1.x, g0.y, g1.y);  // interleave k, k+1
        }
    };
    auto storeB = [&]() {
        #pragma unroll
        for (int i = 0; i < 12; i++) {
            int idx = i * 256 + tid;
            int mat = idx >> 10;
            int rem = idx & 1023;
            int kp = rem >> 7;
            int cp = rem & 127;
            *reinterpret_cast<float4*>(&lB[mat * BMAT + kp * BKP + cp * 4]) = pB[i];
        }
    };

    issueA(0, smem);
    loadB(0);
    int cur = 0;
    for (int k0 = 0; k0 < D2; k0 += KB) {
        storeB();
        async_wait_all();        // own async A(cur) writes landed in LDS
        __syncthreads();         // A(cur) + B(k0) visible to all waves
        if (k0 + KB < D2) {
            issueA(k0 + KB, smem + (cur ^ 1) * LA_FLOATS);  // hidden under WMMAs
            loadB(k0 + KB);
        }
        const float* lAc = smem + cur * LA_FLOATS;

        #pragma unroll
        for (int kk = 0; kk < KB; kk += 4) {
            v2f a_mean = *reinterpret_cast<const v2f*>(
                &lAc[(row_sub * 16 + m16) * A_STRIDE + kk + kb2]);
            v2f a_x = *reinterpret_cast<const v2f*>(
                &lAc[64 * A_STRIDE + (row_sub * 16 + m16) * A_STRIDE + kk + kb2]);
            const int kp = (kk + kb2) >> 1;
            #pragma unroll
            for (int t = 0; t < 8; t++) {
                int col2 = (col_half * 128 + t * 16 + m16) * 2;
                v2f bl_f = *reinterpret_cast<const v2f*>(&lB[0 * BMAT + kp * BKP + col2]);
                v2f br_f = *reinterpret_cast<const v2f*>(&lB[1 * BMAT + kp * BKP + col2]);
                v2f bf_f = *reinterpret_cast<const v2f*>(&lB[2 * BMAT + kp * BKP + col2]);
                acc_g[t] = __builtin_amdgcn_wmma_f32_16x16x4_f32(
                    false, a_mean, false, bl_f, (short)0, acc_g[t], false, false);
                acc_g[t] = __builtin_amdgcn_wmma_f32_16x16x4_f32(
                    false, a_x, false, br_f, (short)0, acc_g[t], false, false);
                acc_m[t] = __builtin_amdgcn_wmma_f32_16x16x4_f32(
                    false, a_x, false, bf_f, (short)0, acc_m[t], false, false);
            }
        }
        __syncthreads();
        cur ^= 1;
    }

    // ---- epilogue: bias + relu + relu-add, store x_combined ----
    #pragma unroll
    for (int t = 0; t < 8; t++) {
        int col = block_col + col_half * 128 + t * 16 + m16;
        float bl_v = bl[col];
        float bf_v = bfc1[col];
        #pragma unroll
        for (int v = 0; v < 8; v++) {
            int m = v + 8 * (lane >> 4);
            int row = block_row + row_sub * 16 + m;
            float g = acc_g[t][v] + bl_v;
            float q = acc_m[t][v] + bf_v;
            g = g > 0.f ? g : 0.f;
            q = q > 0.f ? q : 0.f;
            xcomb[(size_t)row * HID + col] = g + q;
        }
    }
}

// ---------------- segmented per-graph sum (batch is sorted) ----------------
#define NODES_PER_CHUNK 2048
__global__ void graph_reduce(const float* __restrict__ xcomb,
                             const int* __restrict__ batch,
                             float* __restrict__ gsum) {
    int col = blockIdx.x * 256 + threadIdx.x;   // gridDim.x = HID/256
    int n0 = blockIdx.y * NODES_PER_CHUNK;
    int n1 = n0 + NODES_PER_CHUNK;
    float s = 0.0f;
    int cur = batch[n0];
    for (int n = n0; n < n1; n++) {
        int g = batch[n];
        if (g != cur) {
            atomicAdd(&gsum[cur * HID + col], s);
            s = 0.0f;
            cur = g;
        }
        s += xcomb[(size_t)n * HID + col];
    }
    atomicAdd(&gsum[cur * HID + col], s);
}

// ---------------- final head: sigmoid(mean_graph @ W_out + b_out) ----------
__global__ void final_out(const float* __restrict__ gsum,
                          const int* __restrict__ gcnt,
                          const float* __restrict__ Wout,
                          const float* __restrict__ bout,
                          float* __restrict__ out) {
    __shared__ float red[256];
    int g = blockIdx.x;
    int tid = threadIdx.x;
    float s = 0.0f;
    for (int i = tid; i < HID; i += 256) s += gsum[g * HID + i] * Wout[i];
    red[tid] = s;
    __syncthreads();
    for (int off = 128; off > 0; off >>= 1) {
        if (tid < off) red[tid] += red[tid + off];
        __syncthreads();
    }
    if (tid == 0) {
        int c = gcnt[g];
        if (c < 1) c = 1;
        float z = red[0] / (float)c + bout[0];
        out[g] = 1.0f / (1.0f + expf(-z));
    }
}

// ---------------- launcher ----------------
extern "C" void kernel_launch(void* const* d_in, const int* in_sizes, int n_in,
                              void* d_out, int out_size, void* d_ws, size_t ws_size,
                              hipStream_t stream) {
    const float* normal  = (const float*)d_in[0];
    const float* extreme = (const float*)d_in[1];
    const int*   ei      = (const int*)d_in[2];
    const int*   batch   = (const int*)d_in[3];
    const float* Wl      = (const float*)d_in[4];
    const float* bl      = (const float*)d_in[5];
    const float* Wr      = (const float*)d_in[6];
    const float* Wfc1    = (const float*)d_in[7];
    const float* bfc1    = (const float*)d_in[8];
    const float* Wout    = (const float*)d_in[9];
    const float* bout    = (const float*)d_in[10];
    float* out = (float*)d_out;

    char* ws = (char*)d_ws;
    size_t off = 0;
    float* meanBuf = (float*)(ws + off); off += (size_t)N_NODES * D2 * sizeof(float);
    float* xcomb   = (float*)(ws + off); off += (size_t)N_NODES * HID * sizeof(float);
    int* deg       = (int*)(ws + off);   off += (size_t)N_NODES * sizeof(int);
    int* offs      = (int*)(ws + off);   off += (size_t)N_NODES * sizeof(int);
    int* cursor    = (int*)(ws + off);   off += (size_t)N_NODES * sizeof(int);
    int* srcSorted = (int*)(ws + off);   off += (size_t)N_EDGES * sizeof(int);
    float* gsum    = (float*)(ws + off); off += (size_t)N_GRAPHS * HID * sizeof(float);
    int* gcnt      = (int*)(ws + off);   off += (size_t)N_GRAPHS * sizeof(int);

    zero_i32<<<N_NODES / 256, 256, 0, stream>>>(deg, N_NODES);
    zero_f32<<<(N_GRAPHS * HID) / 256, 256, 0, stream>>>(gsum, N_GRAPHS * HID);
    zero_i32<<<1, 256, 0, stream>>>(gcnt, N_GRAPHS);

    edge_degree<<<N_EDGES / 256, 256, 0, stream>>>(ei, deg);
    graph_count<<<N_NODES / 256, 256, 0, stream>>>(batch, gcnt);
    scan_deg<<<1, 256, 0, stream>>>(deg, offs, cursor);
    bin_edges<<<N_EDGES / 256, 256, 0, stream>>>(ei, cursor, srcSorted);
    aggregate_mean<<<N_NODES, 128, 0, stream>>>(normal, extreme, offs, deg,
                                                srcSorted, meanBuf);

    dim3 ggrid(N_NODES / 64, HID / 256);
    fused_gemm<<<ggrid, 256, SMEM_FLOATS * sizeof(float), stream>>>(
        normal, extreme, meanBuf, Wl, bl, Wr, Wfc1, bfc1, xcomb);

    dim3 rgrid(HID / 256, N_NODES / NODES_PER_CHUNK);
    graph_reduce<<<rgrid, 256, 0, stream>>>(xcomb, batch, gsum);

    final_out<<<N_GRAPHS, 256, 0, stream>>>(gsum, gcnt, Wout, bout, out);
}